// YoloV3Loss_89575837926028
// MI455X (gfx1250) — compile-verified
//
#include <hip/hip_runtime.h>

// YOLOv3 loss, decomposed as dense softplus/square channel reductions (15/255
// channels actually read, ~14 MB useful traffic -> latency/bandwidth trivial
// at 23.3 TB/s) + sparse per-target corrections. Dense kernels are templated
// per pyramid level so index math is compile-time mul-shift; per-wave
// reductions use V_WMMA_F32_16X16X4_F32 with a ones B-matrix (exact in f32).

#define B_   64
#define N_   50
#define NCLS 80

typedef float v2f __attribute__((ext_vector_type(2)));
typedef float v8f __attribute__((ext_vector_type(8)));

__constant__ float AW[9] = {116.f/416.f, 156.f/416.f, 373.f/416.f,
                            30.f/416.f,  62.f/416.f,  59.f/416.f,
                            10.f/416.f,  16.f/416.f,  33.f/416.f};
__constant__ float AH[9] = {90.f/416.f, 198.f/416.f, 326.f/416.f,
                            61.f/416.f,  45.f/416.f, 119.f/416.f,
                            13.f/416.f,  30.f/416.f,  23.f/416.f};

__device__ __forceinline__ float softplusf(float z) {
  // stable: log(1+e^z) = log1p(e^{-|z|}) + max(z,0)
  return log1pf(expf(-fabsf(z))) + fmaxf(z, 0.0f);
}

// Exact f32 wave32 reduction via V_WMMA_F32_16X16X4_F32.
// A layout (16x4): lanes 0-15 -> (M=lane,K=0)=v0,(K=1)=v1; lanes 16-31 -> (K=2)=v0,(K=3)=v1.
// With v0 = x, v1 = 0, B = ones: D[m,n] = x[m] + x[m+16].
// C/D layout: VGPR r, lanes 0-15 = row r, lanes 16-31 = row r+8.
// Per-lane sum of the 8 D VGPRs gives half-sums; one shfl_xor(16) completes.
__device__ __forceinline__ float wave_reduce_wmma(float x) {
  v2f a; a[0] = x;    a[1] = 0.0f;
  v2f b; b[0] = 1.0f; b[1] = 1.0f;
  v8f c = {};
  c = __builtin_amdgcn_wmma_f32_16x16x4_f32(false, a, false, b,
                                            (short)0, c, false, false);
  float s = c[0] + c[1] + c[2] + c[3] + c[4] + c[5] + c[6] + c[7];
  s += __shfl_xor(s, 16, 32);
  return s;
}

// Accumulator layout in d_ws (per level L, base = L*16):
//  0:K  1:W  2:corr_x  3:corr_y  4:corr_w  5:corr_h  6:corr_c
//  7:cls_softplus  8:cls_z  9:Sx 10:Sy 11:Sc 12:Sw2 13:Sh2
__global__ void zero_acc(float* __restrict__ acc) {
  if (threadIdx.x < 48) acc[threadIdx.x] = 0.0f;
}

// One block per batch row. Duplicate-scatter resolution ("last n wins", and
// distinct (cell,class) pairs for cls_t) needs only an O(N) scan within the
// row since flat indices encode b.
__global__ void yolo_sparse(const float* __restrict__ p13,
                            const float* __restrict__ p26,
                            const float* __restrict__ p52,
                            const float* __restrict__ tgt,
                            float* __restrict__ acc) {
  __shared__ int s_lvl[N_], s_cell[N_], s_cls[N_];
  const int b = blockIdx.x;
  const int n = threadIdx.x;

  int lvl = -1, cell = -1, cls = -1, a = 0, gx = 0, gy = 0, F = 13;
  float tx = 0.f, ty = 0.f, tw = 0.f, th = 0.f, wh = 0.f;

  if (n < N_) {
    const float* t = tgt + (b * N_ + n) * 5;
    const float cx = t[0], cy = t[1], w = t[2], h = t[3];
    cls = (int)t[4];
    int best = 0; float bi = -1.0f;
    #pragma unroll
    for (int k = 0; k < 9; ++k) {
      float inter = fminf(w, AW[k]) * fminf(h, AH[k]);
      float uni   = w * h + AW[k] * AH[k] - inter;
      float iou   = inter / uni;
      if (iou > bi) { bi = iou; best = k; }   // first max wins (argmax)
    }
    lvl = best / 3;  a = best - lvl * 3;
    F = 13 << lvl;
    float fx = cx * (float)F, fy = cy * (float)F;
    gx = (int)floorf(fx);  gy = (int)floorf(fy);
    tx = fx - (float)gx;   ty = fy - (float)gy;
    tw = logf(w / AW[best]); th = logf(h / AH[best]);
    wh = w * h;
    cell = (a * F + gy) * F + gx;
    s_lvl[n] = lvl; s_cell[n] = cell; s_cls[n] = cls;
  }
  __syncthreads();
  if (n >= N_) return;

  bool winC = true, winCC = true;           // cell winner / (cell,class) rep
  for (int m = n + 1; m < N_; ++m) {
    if (s_lvl[m] == lvl && s_cell[m] == cell) {
      winC = false;
      if (s_cls[m] == cls) winCC = false;
    }
  }
  if (!winC && !winCC) return;

  const float* P = (lvl == 0) ? p13 : ((lvl == 1) ? p26 : p52);
  const int FF   = F * F;
  const int base = ((b * 255 + a * 85) * FF) + gy * F + gx;
  float* A = acc + lvl * 16;

  if (winCC) {                              // every distinct class bit subtracts its logit
    atomicAdd(A + 8, P[base + (5 + cls) * FF]);
  }
  if (winC) {
    float zx = P[base];
    float zy = P[base +     FF];
    float pw = P[base + 2 * FF];
    float ph = P[base + 3 * FF];
    float zc = P[base + 4 * FF];
    atomicAdd(A + 0, 1.0f);
    atomicAdd(A + 1, 2.0f - wh);
    atomicAdd(A + 2, tx * zx);
    atomicAdd(A + 3, ty * zy);
    atomicAdd(A + 4, tw * tw - 2.0f * pw * tw);
    atomicAdd(A + 5, th * th - 2.0f * ph * th);
    atomicAdd(A + 6, zc);
    float sp = 0.0f;
    for (int c2 = 0; c2 < NCLS; ++c2) sp += softplusf(P[base + (5 + c2) * FF]);
    atomicAdd(A + 7, sp);
  }
}

// Dense channel reductions. Templated on level -> FF is a compile-time
// constant, so slice/remainder math lowers to mul-shift (no runtime div in
// the hot loop). gridDim.y = 5 channels; q in {0,1,4}: softplus, {2,3}: square.
template <int LVL>
__global__ void yolo_dense(const float* __restrict__ P,
                           float* __restrict__ acc) {
  constexpr int F  = 13 << LVL;
  constexpr int FF = F * F;
  constexpr int TOTAL = B_ * 3 * FF;
  const int q  = blockIdx.y;
  const bool sq = (q == 2) || (q == 3);

  __shared__ float red[8];

  float s = 0.0f;
  const int stride = gridDim.x * blockDim.x;
  for (int i = blockIdx.x * blockDim.x + threadIdx.x; i < TOTAL; i += stride) {
    int slice = i / FF;           // compile-time const divisor -> mul-shift
    int r     = i - slice * FF;
    int b     = slice / 3;
    int a2    = slice - b * 3;
    float z = P[(b * 255 + a2 * 85 + q) * FF + r];
    s += sq ? z * z : softplusf(z);
  }

  // uniform control flow here: EXEC all ones, as WMMA requires
  float wsum = wave_reduce_wmma(s);
  const int lane = threadIdx.x & 31;
  const int wave = threadIdx.x >> 5;
  if (lane == 0) red[wave] = wsum;
  __syncthreads();
  if (threadIdx.x == 0) {
    float t = 0.0f;
    #pragma unroll
    for (int w = 0; w < 8; ++w) t += red[w];
    int slot = (q == 0) ? 9 : (q == 1) ? 10 : (q == 4) ? 11 : (q == 2) ? 12 : 13;
    atomicAdd(acc + LVL * 16 + slot, t);    // one atomic per block
  }
}

__global__ void yolo_final(const float* __restrict__ acc, float* __restrict__ out) {
  float tot = 0.0f, Ks = 0.0f;
  for (int L = 0; L < 3; ++L) {
    const float* A = acc + L * 16;
    const int F = 13 << L;
    const float size = (float)(B_ * 3 * F * F);
    const float K = A[0], W = A[1];
    float lx = ((A[9]  - A[2]) / size) * W;           // bce_mean(zx,tx)*sum(scale*obj)
    float ly = ((A[10] - A[3]) / size) * W;
    float lw = ((A[12] + A[4]) / size) * 0.5f * W;    // mse * 0.5*sum(scale*obj)
    float lh = ((A[13] + A[5]) / size) * 0.5f * W;
    float lc = A[11] - A[6];                          // bce_c*(sum(obj)+sum(noobj)) = sum BCE
    float lcls = 5.0f * (A[7] - A[8]) / (K * (float)NCLS);
    tot += lx + ly + lw + lh + lc + lcls;
    Ks  += K;
  }
  *out = tot / Ks;
}

extern "C" void kernel_launch(void* const* d_in, const int* in_sizes, int n_in,
                              void* d_out, int out_size, void* d_ws, size_t ws_size,
                              hipStream_t stream) {
  const float* p13 = (const float*)d_in[0];
  const float* p26 = (const float*)d_in[1];
  const float* p52 = (const float*)d_in[2];
  const float* tgt = (const float*)d_in[3];
  float* acc = (float*)d_ws;                // 48 floats used

  zero_acc   <<<1, 64, 0, stream>>>(acc);
  yolo_sparse<<<B_, 64, 0, stream>>>(p13, p26, p52, tgt, acc);
  // Per-level grids sized so each thread handles ~8 elements on a chip
  // this wide; 5 channel-categories in grid.y.
  yolo_dense<0><<<dim3(16,  5), 256, 0, stream>>>(p13, acc);
  yolo_dense<1><<<dim3(64,  5), 256, 0, stream>>>(p26, acc);
  yolo_dense<2><<<dim3(256, 5), 256, 0, stream>>>(p52, acc);
  yolo_final <<<1, 1, 0, stream>>>(acc, (float*)d_out);
}